// ROIPooler_88304527606309
// MI455X (gfx1250) — compile-verified
//
#include <hip/hip_runtime.h>
#include <cstdint>
#include <cstddef>

// ---------------------------------------------------------------------------
// FPN ROIAlign pooler for MI455X (gfx1250).
// Gather-bound op (~800MB L2-resident reads, ~0.4 GFLOP -> 0.0005 FLOP/B):
// WMMA cannot help; the dials are VMEM request count and latency hiding.
//  - CDNA5 async global->LDS meta staging (global_load_async_to_lds_b32 +
//    s_wait_asynccnt) per block.
//  - Fixed-bin thread mapping: all sample indices/weights loop-invariant
//    in registers; zero div/mod or LDS traffic in the hot loop.
//  - Bilinear x-pairs fetched as global_load_b64 (adjacent floats), border
//    clamp folded into pre-adjusted weights -> 8 VMEM ops/channel, not 16.
//    (4B-aligned b64 is legal: ISA 9.5 DWORD mode aligns to min(elem,4)B.)
// ---------------------------------------------------------------------------

#define NP        7                    // pooled size P
#define NC        256                  // channels
#define NROI      1024                 // B*R = 2*512
#define NBIN      (NP * NP)            // 49
#define NQ        5                    // channel residue classes per block
#define NACT      (NQ * NBIN)          // 245 active threads
#define CCHUNK    64                   // channels per block (grid.y = 4)
#define ROISTRIDE (NC * NBIN)          // 12544 outputs per ROI

__device__ __constant__ const int kHt[4] = {200, 100, 50, 25};
__device__ __constant__ const int kWt[4] = {336, 168, 84, 42};

// ---------------------------------------------------------------------------
// Kernel 1: per-ROI level assignment + scaled geometry -> 8 dwords per ROI.
// ---------------------------------------------------------------------------
__global__ void roi_meta_kernel(const float* __restrict__ boxes,
                                float* __restrict__ meta) {
    int n = blockIdx.x * blockDim.x + threadIdx.x;
    if (n >= NROI) return;
    const float* bx = boxes + n * 4;
    float x1 = bx[0], y1 = bx[1], x2 = bx[2], y2 = bx[3];

    float size = sqrtf((x2 - x1) * (y2 - y1));
    float lf = floorf(4.0f + log2f(size / 224.0f + 2.220446049250313e-16f));
    lf = fminf(fmaxf(lf, 2.0f), 5.0f);
    int lvl = (int)lf - 2;                       // 0..3
    float scale = 1.0f / (float)(4 << lvl);      // 0.25 .. 0.03125

    float x1s = x1 * scale, y1s = y1 * scale;
    float rw = fmaxf(x2 * scale - x1s, 1.0f);
    float rh = fmaxf(y2 * scale - y1s, 1.0f);

    float* m = meta + (size_t)n * 8;
    m[0] = __int_as_float(lvl);
    m[1] = __int_as_float(n >> 9);               // batch index (R=512)
    m[2] = x1s;
    m[3] = y1s;
    m[4] = rw * (1.0f / NP);
    m[5] = rh * (1.0f / NP);
    m[6] = 0.0f;
    m[7] = 0.0f;
}

// Reference `prep` + mask fold:
//   mv = (t in [-1,L]); t = max(t,0); lo = floor(t);
//   if lo >= L-1: lo = L-1, f = 0;  w0=(1-f)*mv, w1=f*mv; hi=min(lo+1,L-1)
__device__ __forceinline__ void prep_sample(float t, int L,
                                            int& lo, int& hi,
                                            float& w0, float& w1) {
    float mv = (t >= -1.0f && t <= (float)L) ? 1.0f : 0.0f;
    t = fmaxf(t, 0.0f);
    int l = (int)floorf(t);
    float f;
    if (l >= L - 1) { l = L - 1; f = 0.0f; }
    else            { f = t - (float)l; }
    lo = l;
    hi = min(l + 1, L - 1);
    w0 = (1.0f - f) * mv;
    w1 = f * mv;
}

// ---------------------------------------------------------------------------
// Kernel 2: ROIAlign. grid = (NROI, NC/CCHUNK); block = 256 threads.
// Thread t (< 245): fixed bin r = t%49, channel class q = t/49.
// ---------------------------------------------------------------------------
__global__ __launch_bounds__(256)
void roi_align_kernel(const float* __restrict__ f0,   // x2 [2,256,200,336]
                      const float* __restrict__ f1,   // x3 [2,256,100,168]
                      const float* __restrict__ f2,   // x4 [2,256, 50, 84]
                      const float* __restrict__ f3,   // x5 [2,256, 25, 42]
                      const float* __restrict__ meta,
                      float* __restrict__ out) {
    __shared__ float smeta[8];

    const int roi = blockIdx.x;
    const int tid = threadIdx.x;

    // ---- Stage ROI meta (8 dwords) via CDNA5 async global->LDS DMA --------
    // Flat LDS aperture: addr[31:0] is the in-group LDS offset (ISA 10.2).
    if (tid < 32) {                                  // wave 0 only
        if (tid < 8) {
            uint32_t ldsoff = (uint32_t)(uintptr_t)(&smeta[tid]);
            uint32_t voff   = (uint32_t)(tid * 4);
            const float* sa = meta + (size_t)roi * 8;
            asm volatile("global_load_async_to_lds_b32 %0, %1, %2"
                         :: "v"(ldsoff), "v"(voff), "s"(sa)
                         : "memory");
        }
        asm volatile("s_wait_asynccnt 0" ::: "memory");
    }
    __syncthreads();

    if (tid >= NACT) return;

    const int   lvl = __float_as_int(smeta[0]);
    const int   bb  = __float_as_int(smeta[1]);
    const float x1s = smeta[2];
    const float y1s = smeta[3];
    const float bw  = smeta[4];   // roi_w / P
    const float bh  = smeta[5];   // roi_h / P

    const int H = kHt[lvl];
    const int W = kWt[lvl];
    const float* feat = (lvl == 0) ? f0 : (lvl == 1) ? f1 : (lvl == 2) ? f2 : f3;

    // ---- Per-thread bin + channel class -----------------------------------
    const int q  = tid / NBIN;          // 0..4
    const int r  = tid - q * NBIN;      // 0..48
    const int ph = r / NP;
    const int pw = r - ph * NP;

    // ---- Register-resident sample descriptors -----------------------------
    int yl0, yh0, yl1, yh1, xl0, xh0, xl1, xh1;
    float wy00, wy01, wy10, wy11, u00, u01, u10, u11;
    prep_sample(y1s + ((float)ph + 0.25f) * bh, H, yl0, yh0, wy00, wy01);
    prep_sample(y1s + ((float)ph + 0.75f) * bh, H, yl1, yh1, wy10, wy11);
    prep_sample(x1s + ((float)pw + 0.25f) * bw, W, xl0, xh0, u00, u01);
    prep_sample(x1s + ((float)pw + 0.75f) * bw, W, xl1, xh1, u10, u11);

    // ---- Fold x clamp into weights so each x-pair is ONE b64 load ---------
    // Pair base xs = min(xl, W-2); contribution u0*v[xl] + u1*v[xh]
    // == a0*v[xs] + a1*v[xs+1] with loop-invariant a0,a1.
    const int xs0 = min(xl0, W - 2);
    const int xs1 = min(xl1, W - 2);
    const float a00 = (xl0 == xs0     ? u00 : 0.0f) + (xh0 == xs0     ? u01 : 0.0f);
    const float a01 = (xl0 == xs0 + 1 ? u00 : 0.0f) + (xh0 == xs0 + 1 ? u01 : 0.0f);
    const float a10 = (xl1 == xs1     ? u10 : 0.0f) + (xh1 == xs1     ? u11 : 0.0f);
    const float a11 = (xl1 == xs1 + 1 ? u10 : 0.0f) + (xh1 == xs1 + 1 ? u11 : 0.0f);

    const int   rofs[4] = {yl0 * W, yh0 * W, yl1 * W, yh1 * W};
    const float wy[4]   = {wy00, wy01, wy10, wy11};

    // ---- Channel loop: 8 b64 loads + ~20 FMAs per channel -----------------
    const size_t plane = (size_t)H * (size_t)W;
    const int    cbase = blockIdx.y * CCHUNK;
    const float* fb    = feat + ((size_t)bb * NC + cbase) * plane;
    float*       outp  = out + (size_t)roi * ROISTRIDE + (size_t)cbase * NBIN;

#pragma unroll 2
    for (int c = q; c < CCHUNK; c += NQ) {
        const float* fc = fb + (size_t)c * plane;
        float acc = 0.0f;
#pragma unroll
        for (int a = 0; a < 4; ++a) {
            const float* rp = fc + rofs[a];
            float2 pA = *(const float2*)(rp + xs0);   // b64, 4B-aligned OK
            float2 pB = *(const float2*)(rp + xs1);
            acc += wy[a] * (a00 * pA.x + a01 * pA.y +
                            a10 * pB.x + a11 * pB.y);
        }
        outp[c * NBIN + r] = acc * 0.25f;   // mean over 2x2 samples
    }
}

// ---------------------------------------------------------------------------
extern "C" void kernel_launch(void* const* d_in, const int* in_sizes, int n_in,
                              void* d_out, int out_size, void* d_ws, size_t ws_size,
                              hipStream_t stream) {
    (void)in_sizes; (void)n_in; (void)out_size; (void)ws_size;
    const float* x2    = (const float*)d_in[0];
    const float* x3    = (const float*)d_in[1];
    const float* x4    = (const float*)d_in[2];
    const float* x5    = (const float*)d_in[3];
    const float* boxes = (const float*)d_in[4];
    float* meta = (float*)d_ws;                 // 1024*8 floats = 32 KB
    float* outp = (float*)d_out;

    roi_meta_kernel<<<dim3(NROI / 256), dim3(256), 0, stream>>>(boxes, meta);
    roi_align_kernel<<<dim3(NROI, NC / CCHUNK), dim3(256), 0, stream>>>(
        x2, x3, x4, x5, meta, outp);
}